// QuantumLayer_89661737271308
// MI455X (gfx1250) — compile-verified
//
#include <hip/hip_runtime.h>
#include <math.h>

typedef __attribute__((ext_vector_type(2))) float v2f;
typedef __attribute__((ext_vector_type(8))) float v8f;

#define CUTOFF 32

// ---------------------------------------------------------------------------
// Kernel 1: build G' = kerr-diag * D(gamma)R(phi)S(zeta) Fock matrix (32x32),
// via the exact recurrences, once, on a single workgroup. Output to gbuf:
//   gbuf[0    .. 1023] = Re G'[m][n]  (m-major)
//   gbuf[1024 .. 2047] = Im G'[m][n]
// ---------------------------------------------------------------------------
__global__ void build_g(const float* gamma_re, const float* gamma_im,
                        const float* phi_p, const float* zeta_re,
                        const float* zeta_im, const float* kappa_p,
                        float* __restrict__ gbuf) {
  __shared__ float Gr[CUTOFF][CUTOFF];
  __shared__ float Gi[CUTOFF][CUTOFF];

  if (threadIdx.x == 0) {
    const float gr = *gamma_re, gi = *gamma_im;
    const float ph = *phi_p;
    const float zr = *zeta_re, zi = *zeta_im;

    const float r     = sqrtf(zr * zr + zi * zi);
    const float delta = atan2f(zi, zr);
    const float T     = tanhf(r);
    const float sech  = 1.0f / coshf(r);
    const float adp   = delta + 2.0f * ph;
    const float edr = cosf(adp), edi = sinf(adp);     // e^{i(delta+2phi)}
    const float epr = cosf(ph),  epi = sinf(ph);      // e^{i phi}

    // g00 = exp(-0.5*(|g|^2 + conj(g)^2 * eidp * T)) * sqrt(sech)
    const float cg2r = gr * gr - gi * gi;
    const float cg2i = -2.0f * gr * gi;
    const float wr = (cg2r * edr - cg2i * edi) * T;
    const float wi = (cg2r * edi + cg2i * edr) * T;
    const float exr = -0.5f * ((gr * gr + gi * gi) + wr);
    const float exi = -0.5f * wi;
    const float mag = expf(exr) * sqrtf(sech);
    const float g00r = mag * cosf(exi);
    const float g00i = mag * sinf(exi);

    // A = gamma + conj(gamma)*eidp*T ;  B = eidp*T
    const float Ar = gr + (gr * edr + gi * edi) * T;
    const float Ai = gi + (gr * edi - gi * edr) * T;
    const float Br = edr * T, Bi = edi * T;
    // P = e^{i phi} sech ;  Q = e^{-i delta} T ;  gbar = conj(gamma)
    const float Pr = epr * sech, Pi = epi * sech;
    const float Qr = cosf(delta) * T, Qi = -sinf(delta) * T;
    const float gbr = gr, gbi = -gi;

    float sq[CUTOFF];
    for (int m = 0; m < CUTOFF; ++m) sq[m] = sqrtf((float)m);

    // first column: G[m,0] = (A*G[m-1,0] - B*sqrt(m-1)*G[m-2,0]) / sqrt(m)
    Gr[0][0] = g00r; Gi[0][0] = g00i;
    for (int m = 1; m < CUTOFF; ++m) {
      const float pr = Gr[m - 1][0], pi = Gi[m - 1][0];
      const float qr = (m >= 2) ? Gr[m - 2][0] : 0.0f;
      const float qi = (m >= 2) ? Gi[m - 2][0] : 0.0f;
      const float s = sq[m - 1];
      const float tr = Ar * pr - Ai * pi - s * (Br * qr - Bi * qi);
      const float ti = Ar * pi + Ai * pr - s * (Br * qi + Bi * qr);
      Gr[m][0] = tr / sq[m];
      Gi[m][0] = ti / sq[m];
    }

    // remaining columns:
    // G[m,n] = (P*(sqrt(m)G[m-1,n-1] - gbar*G[m,n-1]) + Q*sqrt(n-1)*G[m,n-2]) / sqrt(n)
    for (int n = 1; n < CUTOFF; ++n) {
      const float inv = 1.0f / sq[n];
      const float sn1 = sq[n - 1];
      for (int m = 0; m < CUTOFF; ++m) {
        const float cr = Gr[m][n - 1], ci = Gi[m][n - 1];
        const float ur = (m >= 1) ? sq[m] * Gr[m - 1][n - 1] : 0.0f;
        const float ui = (m >= 1) ? sq[m] * Gi[m - 1][n - 1] : 0.0f;
        const float upr = ur - (gbr * cr - gbi * ci);
        const float upi = ui - (gbr * ci + gbi * cr);
        const float pvr = (n >= 2) ? Gr[m][n - 2] : 0.0f;
        const float pvi = (n >= 2) ? Gi[m][n - 2] : 0.0f;
        const float tr = Pr * upr - Pi * upi + sn1 * (Qr * pvr - Qi * pvi);
        const float ti = Pr * upi + Pi * upr + sn1 * (Qr * pvi + Qi * pvr);
        Gr[m][n] = tr * inv;
        Gi[m][n] = ti * inv;
      }
    }
  }
  __syncthreads();

  // Fold the Kerr diagonal e^{i*kappa*m^2} into row m and write G' out.
  const float kp = *kappa_p;
  for (int idx = threadIdx.x; idx < CUTOFF * CUTOFF; idx += blockDim.x) {
    const int m = idx >> 5;
    const int n = idx & 31;
    const float ang = kp * (float)(m * m);
    const float kr = cosf(ang), ki = sinf(ang);
    const float a = Gr[m][n], b = Gi[m][n];
    gbuf[idx]        = kr * a - ki * b;
    gbuf[1024 + idx] = kr * b + ki * a;
  }
}

// ---------------------------------------------------------------------------
// Kernel 2: out[b,m] = sum_n G'[m,n] * state[b,n]  (complex), WMMA f32 16x16x4.
// WMMA mapping: M -> batch row (16/tile), N -> output Fock m (2 tiles of 16),
// K -> input Fock n (8 chunks of 4).
// Fragment layout (wave32, f32): lane L = (l = L%16, h = L/16),
//   A[m=l][k=4c+2h+j]  (j = component of float2)
//   B[k=4c+2h+j][n=l]  = G'[n][k]   -> contiguous float2 in row-major G'
//   D VGPR j: element [m = j+8h][n = l]
// ---------------------------------------------------------------------------
__global__ void __launch_bounds__(256) apply_gaussian(
    const float* __restrict__ sre, const float* __restrict__ sim,
    const float* __restrict__ gbuf, float* __restrict__ out, int ntiles) {
  const int lane = threadIdx.x & 31;
  const int l = lane & 15;
  const int h = lane >> 4;
  const int wave = blockIdx.x * (blockDim.x >> 5) + (threadIdx.x >> 5);
  const int nwaves = gridDim.x * (blockDim.x >> 5);

  const float* __restrict__ Gre = gbuf;
  const float* __restrict__ Gim = gbuf + 1024;

  // B fragments for both N-tiles, all 8 K-chunks: held in registers for the
  // whole kernel (8 KB matrix total, L2-resident; loaded once per wave).
  v2f bre[2][8], bim[2][8];
#pragma unroll
  for (int t = 0; t < 2; ++t) {
#pragma unroll
    for (int c = 0; c < 8; ++c) {
      const int off = (t * 16 + l) * 32 + c * 4 + 2 * h;
      bre[t][c] = *(const v2f*)(Gre + off);
      bim[t][c] = *(const v2f*)(Gim + off);
    }
  }

  for (int tile = wave; tile < ntiles; tile += nwaves) {
    const int b0 = tile << 4;
    const float* pre = sre + (size_t)(b0 + l) * 32 + 2 * h;
    const float* pim = sim + (size_t)(b0 + l) * 32 + 2 * h;

    v8f rr0 = {}, rr1 = {};   // sum A_re * B_re
    v8f ii0 = {}, ii1 = {};   // sum A_im * B_im   (subtracted at the end)
    v8f mi0 = {}, mi1 = {};   // imag: chained A_re*B_im + A_im*B_re

#pragma unroll
    for (int c = 0; c < 8; ++c) {
      const v2f ar = *(const v2f*)(pre + c * 4);
      const v2f ai = *(const v2f*)(pim + c * 4);

      rr0 = __builtin_amdgcn_wmma_f32_16x16x4_f32(false, ar, false, bre[0][c], (short)0, rr0, false, false);
      ii0 = __builtin_amdgcn_wmma_f32_16x16x4_f32(false, ai, false, bim[0][c], (short)0, ii0, false, false);
      mi0 = __builtin_amdgcn_wmma_f32_16x16x4_f32(false, ar, false, bim[0][c], (short)0, mi0, false, false);
      mi0 = __builtin_amdgcn_wmma_f32_16x16x4_f32(false, ai, false, bre[0][c], (short)0, mi0, false, false);

      rr1 = __builtin_amdgcn_wmma_f32_16x16x4_f32(false, ar, false, bre[1][c], (short)0, rr1, false, false);
      ii1 = __builtin_amdgcn_wmma_f32_16x16x4_f32(false, ai, false, bim[1][c], (short)0, ii1, false, false);
      mi1 = __builtin_amdgcn_wmma_f32_16x16x4_f32(false, ar, false, bim[1][c], (short)0, mi1, false, false);
      mi1 = __builtin_amdgcn_wmma_f32_16x16x4_f32(false, ai, false, bre[1][c], (short)0, mi1, false, false);
    }

    // Store: out[b][m][{re,im}], b = b0 + j + 8h, m = t*16 + l.
    // float2 {re,im} store -> coalesced 8B/lane along m.
#pragma unroll
    for (int j = 0; j < 8; ++j) {
      const int b = b0 + j + 8 * h;
      float* o = out + ((size_t)b * 32 + l) * 2;
      v2f o0 = { rr0[j] - ii0[j], mi0[j] };
      *(v2f*)(o) = o0;
      v2f o1 = { rr1[j] - ii1[j], mi1[j] };
      *(v2f*)(o + 32) = o1;   // (m+16): +32 floats
    }
  }
}

// ---------------------------------------------------------------------------
extern "C" void kernel_launch(void* const* d_in, const int* in_sizes, int n_in,
                              void* d_out, int out_size, void* d_ws, size_t ws_size,
                              hipStream_t stream) {
  (void)n_in; (void)out_size; (void)ws_size;

  const float* sre = (const float*)d_in[0];
  const float* sim = (const float*)d_in[1];
  float* gbuf = (float*)d_ws;   // 8 KB: Re G' then Im G'

  build_g<<<1, 64, 0, stream>>>((const float*)d_in[2], (const float*)d_in[3],
                                (const float*)d_in[4], (const float*)d_in[5],
                                (const float*)d_in[6], (const float*)d_in[7],
                                gbuf);

  const int batch = in_sizes[0] / CUTOFF;
  const int ntiles = batch >> 4;                 // 16 batch rows per wave-tile
  int blocks = 1024;                             // 8 waves/block -> 8192 waves
  const int maxBlocks = (ntiles + 7) / 8;        // >=1 tile per wave
  if (blocks > maxBlocks) blocks = maxBlocks;
  if (blocks < 1) blocks = 1;

  apply_gaussian<<<blocks, 256, 0, stream>>>(sre, sim, gbuf, (float*)d_out, ntiles);
}